// MyGINConv_68032281968991
// MI455X (gfx1250) — compile-verified
//
#include <hip/hip_runtime.h>

#define N_CH 128

typedef __attribute__((ext_vector_type(2))) float v2f;
typedef __attribute__((ext_vector_type(8))) float v8f;

// ---------------------------------------------------------------------------
// Kernel 1: h = x  (h lives in d_out)
// ---------------------------------------------------------------------------
__global__ void gin_copy_kernel(const float4* __restrict__ x,
                                float4* __restrict__ h, int n4) {
    int i = blockIdx.x * blockDim.x + threadIdx.x;
    if (i < n4) h[i] = x[i];
}

// ---------------------------------------------------------------------------
// Kernel 2: h[dst] += x[src]  (one wave32 per edge, 4 channels per lane)
// ---------------------------------------------------------------------------
__global__ void gin_scatter_kernel(const float* __restrict__ x,
                                   const int* __restrict__ src,
                                   const int* __restrict__ dst,
                                   float* __restrict__ h, int n_edges) {
    int t = blockIdx.x * blockDim.x + threadIdx.x;
    int e    = t >> 5;
    int lane = t & 31;
    if (e >= n_edges) return;
    int s = src[e];
    int d = dst[e];
    const float* xs = x + (size_t)s * N_CH;
    float*       hd = h + (size_t)d * N_CH;
#pragma unroll
    for (int q = 0; q < 4; ++q) {
        atomicAdd(&hd[q * 32 + lane], xs[q * 32 + lane]);
    }
}

// ---------------------------------------------------------------------------
// Kernel 3: in-place  h <- h @ W^T + b  using V_WMMA_F32_16X16X4_F32.
// Each wave owns 16 rows. All A fragments (16x128) are pulled into VGPRs
// first, so overwriting the same rows of h afterwards is safe.
//
// WMMA f32 layouts (wave32):
//   A 16x4 : lane -> M = lane&15 ; VGPR v -> K = 2*(lane>>4) + v
//   B 4x16 : lane -> N = lane&15 ; VGPR v -> K = 2*(lane>>4) + v
//   D 16x16: lane -> N = lane&15 ; VGPR v -> M = v + 8*(lane>>4)
// ---------------------------------------------------------------------------
__global__ void __launch_bounds__(256)
gin_gemm_kernel(float* __restrict__ h, const float* __restrict__ W,
                const float* __restrict__ bias, int n_nodes) {
    const int wave = threadIdx.x >> 5;
    const int lane = threadIdx.x & 31;
    const int m0   = (blockIdx.x * 8 + wave) * 16;
    if (m0 >= n_nodes) return;            // whole-wave exit, EXEC stays uniform

    const int lo = lane & 15;
    const int hi = lane >> 4;

    // Load the full A row-block for this wave: 32 K-fragments of 2 floats.
    int arow_idx = m0 + lo;
    if (arow_idx >= n_nodes) arow_idx = n_nodes - 1;   // clamp (keeps EXEC full)
    const float* arow = h + (size_t)arow_idx * N_CH + 2 * hi;

    v2f a[32];
#pragma unroll
    for (int kb = 0; kb < 32; ++kb) {
        a[kb] = *(const v2f*)(arow + kb * 4);
    }

#pragma unroll
    for (int nt = 0; nt < 8; ++nt) {
        const float* wrow = W + (size_t)(nt * 16 + lo) * N_CH + 2 * hi;
        v8f c = {};
#pragma unroll
        for (int kb = 0; kb < 32; ++kb) {
            v2f bf = *(const v2f*)(wrow + kb * 4);
            // 8 args: (neg_a, A, neg_b, B, c_mod, C, reuse_a, reuse_b)
            c = __builtin_amdgcn_wmma_f32_16x16x4_f32(
                    false, a[kb], false, bf, (short)0, c, false, false);
        }
        const float bb = bias[nt * 16 + lo];
#pragma unroll
        for (int v = 0; v < 8; ++v) {
            const int row = m0 + v + 8 * hi;
            if (row < n_nodes) {
                h[(size_t)row * N_CH + nt * 16 + lo] = c[v] + bb;
            }
        }
    }
}

// ---------------------------------------------------------------------------
// Host launcher
// ---------------------------------------------------------------------------
extern "C" void kernel_launch(void* const* d_in, const int* in_sizes, int n_in,
                              void* d_out, int out_size, void* d_ws, size_t ws_size,
                              hipStream_t stream) {
    const float* x    = (const float*)d_in[0];
    const int*   ei   = (const int*)d_in[1];   // [2, E] row-major
    const float* W    = (const float*)d_in[2];
    const float* bias = (const float*)d_in[3];
    float*       h    = (float*)d_out;

    const int n_nodes = in_sizes[0] / N_CH;
    const int n_edges = in_sizes[1] / 2;
    const int* src = ei;
    const int* dst = ei + n_edges;

    // 1) h = x
    {
        int n4 = n_nodes * (N_CH / 4);
        int blocks = (n4 + 255) / 256;
        gin_copy_kernel<<<blocks, 256, 0, stream>>>((const float4*)x,
                                                    (float4*)h, n4);
    }

    // 2) h[dst] += x[src]
    {
        long long threads = (long long)n_edges * 32;
        int blocks = (int)((threads + 255) / 256);
        gin_scatter_kernel<<<blocks, 256, 0, stream>>>(x, src, dst, h, n_edges);
    }

    // 3) h = h @ W^T + b   (in place, WMMA)
    {
        int blocks = (n_nodes + 127) / 128;   // 8 waves x 16 rows per block
        gin_gemm_kernel<<<blocks, 256, 0, stream>>>(h, W, bias, n_nodes);
    }
}